// Graph_Encoder_82686710383106
// MI455X (gfx1250) — compile-verified
//
#include <hip/hip_runtime.h>

typedef __attribute__((ext_vector_type(2))) float v2f;
typedef __attribute__((ext_vector_type(8))) float v8f;

#define N_NODES 50000
#define N_EDGES 1250000
#define DIM     64
#define TILES_M (N_NODES / 16)   // 3125 (exact)
#define TILES_N (DIM / 16)       // 4
#define N_WAVES (TILES_M * TILES_N)

// --- Kernel 1: deg[i] = 1.0f (self-loop weight). ws is not re-poisoned -> re-init each call.
__global__ void k_init_deg(float* __restrict__ deg) {
    int i = blockIdx.x * blockDim.x + threadIdx.x;
    if (i < N_NODES) deg[i] = 1.0f;
}

// --- Kernel 2: weighted in-degree: deg[dst[e]] += ew[e]
__global__ void k_deg(const int* __restrict__ dst, const float* __restrict__ ew,
                      float* __restrict__ deg) {
    int e = blockIdx.x * blockDim.x + threadIdx.x;
    if (e < N_EDGES) atomicAdd(&deg[dst[e]], ew[e]);
}

// --- Kernel 3: dinv = deg > 0 ? rsqrt(deg) : 0
__global__ void k_dinv(const float* __restrict__ deg, float* __restrict__ dinv) {
    int i = blockIdx.x * blockDim.x + threadIdx.x;
    if (i < N_NODES) {
        float d = deg[i];
        dinv[i] = (d > 0.0f) ? __frsqrt_rn(d) : 0.0f;
    }
}

// --- Kernel 4: XW via V_WMMA_F32_16X16X4_F32; also seed out = dinv^2 * xw + bias.
// One wave computes one 16x16 tile of XW. 16 WMMA ops (K = 64, step 4).
__global__ void __launch_bounds__(256)
k_gemm_wmma(const float* __restrict__ x, const float* __restrict__ W,
            const float* __restrict__ bias, const float* __restrict__ dinv,
            float* __restrict__ xw, float* __restrict__ out) {
    const int lane  = threadIdx.x & 31;
    const int gwave = blockIdx.x * (blockDim.x >> 5) + (threadIdx.x >> 5);
    if (gwave >= N_WAVES) return;          // wave-uniform guard: EXEC stays all-1s

    const int tm = gwave / TILES_N;
    const int tn = gwave - tm * TILES_N;
    const int m0 = tm * 16;
    const int n0 = tn * 16;

    const int half = lane >> 4;            // 0: K pair {0,1}, 1: K pair {2,3}
    const int lid  = lane & 15;            // A: row within tile; B/D: column

    const float* __restrict__ arow = x + (size_t)(m0 + lid) * DIM;

    v8f acc = {};
#pragma unroll
    for (int k0 = 0; k0 < DIM; k0 += 4) {
        const int ka = k0 + 2 * half;
        v2f a, b;
        a.x = arow[ka];
        a.y = arow[ka + 1];
        b.x = W[(size_t)ka       * DIM + n0 + lid];
        b.y = W[(size_t)(ka + 1) * DIM + n0 + lid];
        // D = A*B + C   (v_wmma_f32_16x16x4_f32)
        acc = __builtin_amdgcn_wmma_f32_16x16x4_f32(
            /*neg_a=*/false, a, /*neg_b=*/false, b,
            /*c_mod=*/(short)0, acc, /*reuse_a=*/false, /*reuse_b=*/false);
    }

    // D layout: VGPR j, lanes 0-15 -> M=j, lanes 16-31 -> M=j+8; N = lid.
    const int   col = n0 + lid;
    const float bv  = bias[col];
#pragma unroll
    for (int j = 0; j < 8; ++j) {
        const int    row  = m0 + j + 8 * half;
        const size_t idx  = (size_t)row * DIM + col;
        const float  v    = acc[j];
        const float  di   = dinv[row];
        xw[idx]  = v;                      // keep XW for the edge gather
        out[idx] = di * di * v + bv;       // self-loop term + bias
    }
}

// --- Kernel 5: edge scatter: out[dst] += norm * xw[src]  (64 dims per edge)
__global__ void k_scatter(const int* __restrict__ src, const int* __restrict__ dst,
                          const float* __restrict__ ew, const float* __restrict__ dinv,
                          const float* __restrict__ xw, float* __restrict__ out) {
    const unsigned gid = blockIdx.x * blockDim.x + threadIdx.x;   // < E*64 = 80M
    if (gid >= (unsigned)N_EDGES * DIM) return;
    const int e = (int)(gid >> 6);
    const int d = (int)(gid & 63);
    const int s = src[e];
    const int t = dst[e];
    const float norm = dinv[s] * ew[e] * dinv[t];
    atomicAdd(&out[(size_t)t * DIM + d], xw[(size_t)s * DIM + d] * norm);
}

// --- Kernel 6: ReLU
__global__ void k_relu(float* __restrict__ out) {
    int i = blockIdx.x * blockDim.x + threadIdx.x;
    if (i < N_NODES * DIM) out[i] = fmaxf(out[i], 0.0f);
}

extern "C" void kernel_launch(void* const* d_in, const int* in_sizes, int n_in,
                              void* d_out, int out_size, void* d_ws, size_t ws_size,
                              hipStream_t stream) {
    const float* x   = (const float*)d_in[0];            // [N, 64]
    const int*   ei  = (const int*)d_in[1];              // [2, E] flat: src then dst
    const float* ew  = (const float*)d_in[2];            // [E]
    const float* W   = (const float*)d_in[3];            // [64, 64]
    const float* b   = (const float*)d_in[4];            // [64]
    float*       out = (float*)d_out;                    // [N, 64]

    const int* src = ei;
    const int* dst = ei + N_EDGES;

    // Workspace layout: deg[N] | dinv[N] | xw[N*64]  -> 13.2 MB total
    float* deg  = (float*)d_ws;
    float* dinv = deg + N_NODES;
    float* xw   = dinv + N_NODES;

    // 1) deg = 1.0 (self loops)
    k_init_deg<<<(N_NODES + 255) / 256, 256, 0, stream>>>(deg);
    // 2) deg[dst] += ew
    k_deg<<<(N_EDGES + 255) / 256, 256, 0, stream>>>(dst, ew, deg);
    // 3) dinv = rsqrt(deg)
    k_dinv<<<(N_NODES + 255) / 256, 256, 0, stream>>>(deg, dinv);
    // 4) xw = x @ W (WMMA); out = dinv^2*xw + b
    {
        const int waves_per_block = 256 / 32;
        const int blocks = (N_WAVES + waves_per_block - 1) / waves_per_block;
        k_gemm_wmma<<<blocks, 256, 0, stream>>>(x, W, b, dinv, xw, out);
    }
    // 5) edge scatter-add
    {
        const unsigned total = (unsigned)N_EDGES * DIM;
        k_scatter<<<(total + 255) / 256, 256, 0, stream>>>(src, dst, ew, dinv, xw, out);
    }
    // 6) ReLU
    k_relu<<<(N_NODES * DIM + 255) / 256, 256, 0, stream>>>(out);
}